// MhcProjector_83004537963027
// MI455X (gfx1250) — compile-verified
//
#include <hip/hip_runtime.h>
#include <math.h>

// ---------------------------------------------------------------------------
// MhcProjector fused kernel for gfx1250 (MI455X).
//   x: (8192, 4096) f32  ->  rmsnorm -> @ [phi_pre|phi_post|phi_res] (4096x24)
//   -> sigmoid / 2*sigmoid / sinkhorn(4x4, 20 iters)
// Memory-bound: 128 MB of x at 23.3 TB/s ~ 5.5us floor. GEMM via
// V_WMMA_F32_16X16X4_F32 (f32 A/B/C), rms_weight folded into B.
// ---------------------------------------------------------------------------

typedef __attribute__((ext_vector_type(2))) float v2f;
typedef __attribute__((ext_vector_type(8))) float v8f;

#define NROWS 8192   // B*T
#define KDIM  4096   // FLAT
#define NCOL  24     // 4 (pre) + 4 (post) + 16 (res)
#define NPAD  32     // padded B width (two 16-wide WMMA tiles)
#define TMAX  20
#define EPS   1e-6f

// phi element (k, n) of the concatenated 4096 x 24 weight block
__device__ __forceinline__ float phi_at(const float* __restrict__ pre,
                                        const float* __restrict__ post,
                                        const float* __restrict__ res,
                                        int k, int n) {
    if (n < 4)  return pre [k * 4  + n];
    if (n < 8)  return post[k * 4  + (n - 4)];
    if (n < 24) return res [k * 16 + (n - 8)];
    return 0.0f;
}

// Build Bp[kp][n] = ( w[2kp]*phi(2kp,n), w[2kp+1]*phi(2kp+1,n) )
// Interleaved K-pairs so one b64 load per lane yields the WMMA B fragment:
//   VGPR0 = rows {k0, k0+2} across lane halves, VGPR1 = rows {k0+1, k0+3}.
__global__ void mhc_prep_kernel(const float* __restrict__ w,
                                const float* __restrict__ pre,
                                const float* __restrict__ post,
                                const float* __restrict__ res,
                                float2* __restrict__ Bp) {
    int idx = blockIdx.x * blockDim.x + threadIdx.x;
    if (idx >= (KDIM / 2) * NPAD) return;
    int kp = idx >> 5;
    int n  = idx & 31;
    int k0 = kp * 2;
    float2 v;
    v.x = phi_at(pre, post, res, k0,     n) * w[k0];
    v.y = phi_at(pre, post, res, k0 + 1, n) * w[k0 + 1];
    Bp[idx] = v;
}

__global__ __launch_bounds__(256) void mhc_main_kernel(
    const float*  __restrict__ x,
    const float2* __restrict__ Bp,
    const float*  __restrict__ b_pre,
    const float*  __restrict__ b_post,
    const float*  __restrict__ b_res,
    const float*  __restrict__ a_pre,
    const float*  __restrict__ a_post,
    const float*  __restrict__ a_res,
    float* __restrict__ out)
{
    __shared__ float tile[8][16 * NCOL];   // 12 KB: per-wave 16x24 result tile

    const int lane    = threadIdx.x & 31;
    const int wave    = threadIdx.x >> 5;
    const int mrow    = lane & 15;         // row within 16-row tile / B column
    const int khalf   = lane >> 4;         // which K-pair half this lane owns
    const int rowBase = (blockIdx.x * 8 + wave) * 16;

    // A fragment source: lane L<16 reads (k0, k0+1) of row L; L>=16 reads (k0+2, k0+3).
    const float* aptr = x + (size_t)(rowBase + mrow) * KDIM + khalf * 2;

    v8f acc0 = {};   // N = 0..15  (pre 0-3, post 4-7, res cols 0-7)
    v8f acc1 = {};   // N = 16..31 (res cols 8-15, pad 24-31)
    float ss = 0.0f; // partial sum of squares for this lane's row

    #pragma unroll 4
    for (int k0 = 0; k0 < KDIM; k0 += 4) {
        v2f a = *(const v2f*)(aptr + k0);
        ss = fmaf(a.x, a.x, ss);
        ss = fmaf(a.y, a.y, ss);

        const int kp = (k0 >> 1) + khalf;
        const float2* brow = Bp + (size_t)kp * NPAD;
        v2f b0 = *(const v2f*)(brow + mrow);
        v2f b1 = *(const v2f*)(brow + 16 + mrow);

        acc0 = __builtin_amdgcn_wmma_f32_16x16x4_f32(false, a, false, b0,
                                                     (short)0, acc0, false, false);
        acc1 = __builtin_amdgcn_wmma_f32_16x16x4_f32(false, a, false, b1,
                                                     (short)0, acc1, false, false);
    }

    // Lane L and L^16 together cover all K of row (L&15)
    float ssr = ss + __shfl_xor(ss, 16, 32);
    float r   = rsqrtf(ssr * (1.0f / (float)KDIM) + EPS);   // rms scale, rows scaled post-GEMM

    // Spill D to LDS. D layout: element i of acc -> row i+8*khalf, col (lane&15)+16*tileidx
    float* my = &tile[wave][0];
    #pragma unroll
    for (int i = 0; i < 8; ++i) {
        int m = i + 8 * khalf;
        my[m * NCOL + mrow] = acc0[i];
    }
    if (mrow < 8) {   // cols 24..31 of tile1 are padding; keep only 16..23
        #pragma unroll
        for (int i = 0; i < 8; ++i) {
            int m = i + 8 * khalf;
            my[m * NCOL + 16 + mrow] = acc1[i];
        }
    }
    __syncthreads();

    // Epilogue: lanes 0..15 each own one row (r for row j already lives in lane j)
    if (lane < 16) {
        const int row = rowBase + mrow;
        const float s_pre  = a_pre[0]  * r;
        const float s_post = a_post[0] * r;
        const float s_res  = a_res[0]  * r;
        const float* d = &tile[wave][mrow * NCOL];

        float4 o_pre, o_post;
        #pragma unroll
        for (int j = 0; j < 4; ++j) {
            float tp = s_pre  * d[j]     + b_pre[j];
            float tq = s_post * d[4 + j] + b_post[j];
            ((float*)&o_pre)[j]  = 1.0f / (1.0f + __expf(-tp));
            ((float*)&o_post)[j] = 2.0f / (1.0f + __expf(-tq));
        }
        ((float4*)out)[row]                 = o_pre;
        ((float4*)(out + NROWS * 4))[row]   = o_post;

        // Sinkhorn-Knopp on the 4x4, fully in registers
        float M[16];
        #pragma unroll
        for (int e = 0; e < 16; ++e)
            M[e] = __expf(s_res * d[8 + e] + b_res[e]);

        for (int it = 0; it < TMAX; ++it) {
            #pragma unroll
            for (int a2 = 0; a2 < 4; ++a2) {      // row normalize
                float s = M[a2*4] + M[a2*4+1] + M[a2*4+2] + M[a2*4+3];
                float inv = 1.0f / s;
                M[a2*4]   *= inv; M[a2*4+1] *= inv;
                M[a2*4+2] *= inv; M[a2*4+3] *= inv;
            }
            #pragma unroll
            for (int b2 = 0; b2 < 4; ++b2) {      // col normalize
                float s = M[b2] + M[b2+4] + M[b2+8] + M[b2+12];
                float inv = 1.0f / s;
                M[b2]    *= inv; M[b2+4]  *= inv;
                M[b2+8]  *= inv; M[b2+12] *= inv;
            }
        }

        float* outr = out + NROWS * 8 + (size_t)row * 16;
        #pragma unroll
        for (int e = 0; e < 16; e += 4) {
            float4 v = make_float4(M[e], M[e+1], M[e+2], M[e+3]);
            *(float4*)(outr + e) = v;
        }
    }
}

extern "C" void kernel_launch(void* const* d_in, const int* in_sizes, int n_in,
                              void* d_out, int out_size, void* d_ws, size_t ws_size,
                              hipStream_t stream) {
    const float* x      = (const float*)d_in[0];   // (4,2048,4,1024)
    const float* w      = (const float*)d_in[1];   // (4096,)
    const float* ppre   = (const float*)d_in[2];   // (4096,4)
    const float* ppost  = (const float*)d_in[3];   // (4096,4)
    const float* pres   = (const float*)d_in[4];   // (4096,16)
    const float* b_pre  = (const float*)d_in[5];   // (4,)
    const float* b_post = (const float*)d_in[6];   // (4,)
    const float* b_res  = (const float*)d_in[7];   // (4,4)
    const float* a_pre  = (const float*)d_in[8];   // scalar
    const float* a_post = (const float*)d_in[9];   // scalar
    const float* a_res  = (const float*)d_in[10];  // scalar

    float2* Bp = (float2*)d_ws;                    // 2048*32 float2 = 512 KB

    int prep_elems = (KDIM / 2) * NPAD;
    mhc_prep_kernel<<<(prep_elems + 255) / 256, 256, 0, stream>>>(w, ppre, ppost, pres, Bp);

    // 8 waves/block * 16 rows/wave = 128 rows/block -> 64 blocks
    mhc_main_kernel<<<NROWS / 128, 256, 0, stream>>>(
        x, Bp, b_pre, b_post, b_res, a_pre, a_post, a_res, (float*)d_out);
}